// SAGE_79972291051913
// MI455X (gfx1250) — compile-verified
//
#include <hip/hip_runtime.h>
#include <hip/hip_bf16.h>
#include <math.h>

// ---------------------------------------------------------------------------
// GraphSAGE (pool) 2-layer forward for MI455X (gfx1250, wave32, WMMA).
// All GEMMs use V_WMMA_F32_16X16X4_F32 (fp32 in/out): workload is HBM-bound
// (~0.8 GB traffic @ 23.3 TB/s >> 58 GFLOP of fp32 matrix work), so full
// precision is free.
// B tile is stored pair-interleaved in LDS so each WMMA B fragment is a
// single aligned ds_load_b64; fragments for two k-steps are batched ahead
// of the 8 WMMAs to let the scheduler overlap DS latency with matrix ops.
// ---------------------------------------------------------------------------

typedef __attribute__((ext_vector_type(2))) float v2f;
typedef __attribute__((ext_vector_type(8))) float v8f;

#define BM 64
#define BN 64
#define BKC 32
#define SA (BKC + 2)          // 34 floats = 136 B row stride (8B aligned)
#define PITCHW (2 * BN + 2)   // 130 floats = 520 B per k-pair row (8B aligned)

// Generic tiled GEMM:  C[M,N] = act( A1[g(m),K1] @ W1 + A2[m,K2] @ W2 + bias )
// - gidx: optional row gather for A1 (edge_src); A2 never gathered.
// - Grid: (M/64, N/64). Block: 128 threads = 4 waves; wave w owns rows
//   [w*16, w*16+16) of the block tile and all 64 N columns (4 WMMA col tiles).
__global__ __launch_bounds__(128) void gemm_wmma_f32(
    const float* __restrict__ A1, const float* __restrict__ W1, int K1,
    const float* __restrict__ A2, const float* __restrict__ W2, int K2,
    const int* __restrict__ gidx,
    const float* __restrict__ bias,
    float* __restrict__ C, int N, int relu)
{
    __shared__ float lsA[BM * SA];
    __shared__ float lsW[(BKC / 2) * PITCHW];

    const int tid  = threadIdx.x;
    const int wave = tid >> 5;
    const int lane = tid & 31;
    const int m0   = blockIdx.x * BM;
    const int n0   = blockIdx.y * BN;

    v8f acc[4] = {};

    const int mrow  = (wave << 4) + (lane & 15); // A-frag row within block tile
    const int khp   = lane >> 4;                 // lanes 16-31 take k-pair +1

    for (int seg = 0; seg < 2; ++seg) {
        const float* A = seg ? A2 : A1;
        const float* W = seg ? W2 : W1;
        const int    K = seg ? K2 : K1;
        const int* gi  = seg ? nullptr : gidx;
        if (K == 0 || A == nullptr) continue;

        for (int kc = 0; kc < K; kc += BKC) {
            __syncthreads();
            // Stage A tile (BM x BKC), optionally gathered rows.
            for (int i = tid; i < BM * BKC; i += 128) {
                int r = i >> 5, c = i & (BKC - 1);
                int grow = m0 + r;
                if (gi) grow = gi[grow];
                lsA[r * SA + c] = A[(size_t)grow * K + kc + c];
            }
            // Stage W tile (BKC x BN), pair-interleaved over K:
            // lsW[(r/2)*PITCHW + c*2 + (r&1)] = W[kc+r][n0+c]
            for (int i = tid; i < BKC * BN; i += 128) {
                int r = i >> 6, c = i & (BN - 1);
                lsW[(r >> 1) * PITCHW + (c << 1) + (r & 1)] =
                    W[(size_t)(kc + r) * N + n0 + c];
            }
            __syncthreads();

            // Two k-steps per iteration: batch all DS loads, then 8 WMMAs.
            #pragma unroll
            for (int k4 = 0; k4 < BKC; k4 += 8) {
                const int kp0 = (k4 >> 1) + khp;          // k-pair index
                const int kp1 = kp0 + 2;
                const v2f a0 = *reinterpret_cast<const v2f*>(
                    &lsA[mrow * SA + (kp0 << 1)]);
                const v2f a1 = *reinterpret_cast<const v2f*>(
                    &lsA[mrow * SA + (kp1 << 1)]);
                const float* wp0 = &lsW[kp0 * PITCHW + ((lane & 15) << 1)];
                const float* wp1 = &lsW[kp1 * PITCHW + ((lane & 15) << 1)];
                v2f b0[4], b1[4];
                #pragma unroll
                for (int nt = 0; nt < 4; ++nt)
                    b0[nt] = *reinterpret_cast<const v2f*>(wp0 + nt * 32);
                #pragma unroll
                for (int nt = 0; nt < 4; ++nt)
                    b1[nt] = *reinterpret_cast<const v2f*>(wp1 + nt * 32);
                #pragma unroll
                for (int nt = 0; nt < 4; ++nt)
                    acc[nt] = __builtin_amdgcn_wmma_f32_16x16x4_f32(
                        false, a0, false, b0[nt], (short)0, acc[nt], false, false);
                #pragma unroll
                for (int nt = 0; nt < 4; ++nt)
                    acc[nt] = __builtin_amdgcn_wmma_f32_16x16x4_f32(
                        false, a1, false, b1[nt], (short)0, acc[nt], false, false);
            }
        }
    }

    // Epilogue. C layout: VGPR r -> row r (lanes 0-15) / r+8 (lanes 16-31),
    // col = nt*16 + lane%16.
    const int mbase = m0 + (wave << 4) + ((lane >> 4) << 3);
    const int col0  = n0 + (lane & 15);
    #pragma unroll
    for (int nt = 0; nt < 4; ++nt) {
        const int gcol = col0 + nt * 16;
        const float bcol = bias ? bias[gcol] : 0.0f;
        const v8f av = acc[nt];
        #pragma unroll
        for (int r = 0; r < 8; ++r) {
            float v = av[r] + bcol;
            if (relu) v = fmaxf(v, 0.0f);
            C[(size_t)(mbase + r) * N + gcol] = v;
        }
    }
}

// pooled[v,f] = max_{i<fan} m[v*fan+i, f]   (edge_dst is repeat(arange,fan))
__global__ void segmax_kernel(const float* __restrict__ m,
                              float* __restrict__ pooled,
                              int V, int d, int fan)
{
    int i = blockIdx.x * blockDim.x + threadIdx.x;
    int vec = d >> 2;
    if (i >= V * vec) return;
    int v = i / vec;
    int f = (i - v * vec) << 2;
    const float4* mp = reinterpret_cast<const float4*>(m + (size_t)v * fan * d + f);
    float4 best = mp[0];
    for (int e = 1; e < fan; ++e) {
        float4 t = *reinterpret_cast<const float4*>(m + ((size_t)v * fan + e) * d + f);
        best.x = fmaxf(best.x, t.x); best.y = fmaxf(best.y, t.y);
        best.z = fmaxf(best.z, t.z); best.w = fmaxf(best.w, t.w);
    }
    *reinterpret_cast<float4*>(pooled + (size_t)v * d + f) = best;
}

// Row L2 normalization in place; one block (of d threads) per row.
__global__ void l2norm_kernel(float* __restrict__ h, int d)
{
    __shared__ float red[256];
    const int tid = threadIdx.x;
    const size_t row = blockIdx.x;
    float v = h[row * d + tid];
    red[tid] = v * v;
    for (int s = blockDim.x >> 1; s > 0; s >>= 1) {
        __syncthreads();
        if (tid < s) red[tid] += red[tid + s];
    }
    __syncthreads();
    float inv = 1.0f / fmaxf(sqrtf(red[0]), 1e-12f);
    h[row * d + tid] = v * inv;
}

// Deterministic column stats, stage 1: per-chunk sums (no atomics).
#define STAT_ROWS 512
__global__ void colstat_partial(const float* __restrict__ h,
                                float* __restrict__ part, int d)
{
    const int col = threadIdx.x;
    const size_t r0 = (size_t)blockIdx.x * STAT_ROWS;
    float s = 0.0f, ss = 0.0f;
    for (int r = 0; r < STAT_ROWS; ++r) {
        float v = h[(r0 + r) * d + col];
        s += v; ss += v * v;
    }
    part[(size_t)blockIdx.x * 2 * d + col]     = s;
    part[(size_t)blockIdx.x * 2 * d + d + col] = ss;
}

// Stage 2: fold chunks in fixed order; emit scale/shift for BN affine.
__global__ void colstat_finalize(const float* __restrict__ part, int nchunks,
                                 const float* __restrict__ g,
                                 const float* __restrict__ be,
                                 float nrows, int d,
                                 float* __restrict__ ss_out)
{
    const int col = threadIdx.x;
    float s = 0.0f, ss = 0.0f;
    for (int c = 0; c < nchunks; ++c) {
        s  += part[(size_t)c * 2 * d + col];
        ss += part[(size_t)c * 2 * d + d + col];
    }
    float mu  = s / nrows;
    float var = ss / nrows - mu * mu;
    float sc  = g[col] * rsqrtf(var + 1e-5f);
    ss_out[col]     = sc;
    ss_out[d + col] = be[col] - mu * sc;
}

// y = x*scale[col] + shift[col], optional ReLU. d must be 256 here.
__global__ void bn_apply(const float* __restrict__ in, float* __restrict__ out,
                         const float* __restrict__ ss, size_t total, int relu)
{
    size_t i = (size_t)blockIdx.x * blockDim.x + threadIdx.x;
    if (i >= total) return;
    int col = (int)(i & 255);
    float v = in[i] * ss[col] + ss[256 + col];
    if (relu) v = fmaxf(v, 0.0f);
    out[i] = v;
}

// ---------------------------------------------------------------------------

extern "C" void kernel_launch(void* const* d_in, const int* in_sizes, int n_in,
                              void* d_out, int out_size, void* d_ws, size_t ws_size,
                              hipStream_t stream)
{
    (void)in_sizes; (void)n_in; (void)out_size; (void)ws_size;

    constexpr int N1c = 81920, N2c = 8192, FAN0 = 8, FAN1 = 10;
    constexpr int E0 = N1c * FAN0, E1 = N2c * FAN1;
    constexpr int F = 128, H = 256;

    const float* x        = (const float*)d_in[0];
    const int*   esrc0    = (const int*)  d_in[1];
    const int*   esrc1    = (const int*)  d_in[3];
    const float* W_pool0  = (const float*)d_in[5];
    const float* b_pool0  = (const float*)d_in[6];
    const float* W_self0  = (const float*)d_in[7];
    const float* W_neigh0 = (const float*)d_in[8];
    const float* b0       = (const float*)d_in[9];
    const float* W_pool1  = (const float*)d_in[10];
    const float* b_pool1  = (const float*)d_in[11];
    const float* W_self1  = (const float*)d_in[12];
    const float* W_neigh1 = (const float*)d_in[13];
    const float* b1       = (const float*)d_in[14];
    const float* g_bn0    = (const float*)d_in[15];
    const float* be_bn0   = (const float*)d_in[16];
    const float* W_h      = (const float*)d_in[17];
    const float* b_h      = (const float*)d_in[18];
    const float* g_bnh    = (const float*)d_in[19];
    const float* be_bnh   = (const float*)d_in[20];
    const float* W_lat    = (const float*)d_in[21];
    const float* b_lat    = (const float*)d_in[22];
    const float* g_bnl    = (const float*)d_in[23];
    const float* be_bnl   = (const float*)d_in[24];

    float* ws = (float*)d_ws;
    // Workspace layout (floats), regions reused across phases; peak ~210 MB.
    const size_t O_POOLED0 = 0;                               // 10.5M  [P1..P2]
    const size_t O_MCHUNK  = (size_t)N1c * F;                 // 10.5M  [P1 only]
    const size_t O_H0      = O_MCHUNK;                        // 21.0M  [P2..P3]
    const size_t O_H1      = O_H0 + (size_t)N1c * H;          // 21.0M  [P3..P5]
    const size_t O_MEDGE1  = O_H0;                            // 21.0M  [P4] (h0 dead)
    const size_t O_POOLED1 = 0;                               // 2.1M   [P4..P5] (pooled0 dead)
    const size_t O_H2      = (size_t)N2c * H;                 // 2.1M   [P5..P6]
    const size_t O_HL      = O_H2 + (size_t)N2c * H;          // 2.1M   [P6]
    const size_t O_PART    = O_HL + (size_t)N2c * H;          // 82K
    const size_t O_SS      = O_PART + 160 * 2 * (size_t)H;    // 512

    dim3 blk128(128), blk256(256);

    // ---- P1: layer-0 edge messages + pooling, chunked (8 chunks) ----
    constexpr int NCH0 = 8;
    constexpr int ECH0 = E0 / NCH0;    // 81920 edges
    constexpr int VCH0 = N1c / NCH0;   // 10240 dsts
    for (int c = 0; c < NCH0; ++c) {
        gemm_wmma_f32<<<dim3(ECH0 / BM, F / BN), blk128, 0, stream>>>(
            x, W_pool0, F, nullptr, nullptr, 0,
            esrc0 + (size_t)c * ECH0, b_pool0,
            ws + O_MCHUNK, F, 1);
        int tot = VCH0 * (F / 4);
        segmax_kernel<<<(tot + 255) / 256, blk256, 0, stream>>>(
            ws + O_MCHUNK, ws + O_POOLED0 + (size_t)c * VCH0 * F, VCH0, F, FAN0);
    }

    // ---- P2: h0 = x[:N1]@W_self0 + pooled0@W_neigh0 + b0 ----
    gemm_wmma_f32<<<dim3(N1c / BM, H / BN), blk128, 0, stream>>>(
        x, W_self0, F, ws + O_POOLED0, W_neigh0, F,
        nullptr, b0, ws + O_H0, H, 0);

    // ---- P3: l2norm -> BN0+relu -> W_h -> BNh+relu ----
    l2norm_kernel<<<N1c, dim3(H), 0, stream>>>(ws + O_H0, H);
    colstat_partial<<<N1c / STAT_ROWS, blk256, 0, stream>>>(ws + O_H0, ws + O_PART, H);
    colstat_finalize<<<1, blk256, 0, stream>>>(ws + O_PART, N1c / STAT_ROWS,
                                               g_bn0, be_bn0, (float)N1c, H, ws + O_SS);
    bn_apply<<<(unsigned)(((size_t)N1c * H + 255) / 256), blk256, 0, stream>>>(
        ws + O_H0, ws + O_H0, ws + O_SS, (size_t)N1c * H, 1);

    gemm_wmma_f32<<<dim3(N1c / BM, H / BN), blk128, 0, stream>>>(
        ws + O_H0, W_h, H, nullptr, nullptr, 0, nullptr, b_h, ws + O_H1, H, 0);
    colstat_partial<<<N1c / STAT_ROWS, blk256, 0, stream>>>(ws + O_H1, ws + O_PART, H);
    colstat_finalize<<<1, blk256, 0, stream>>>(ws + O_PART, N1c / STAT_ROWS,
                                               g_bnh, be_bnh, (float)N1c, H, ws + O_SS);
    bn_apply<<<(unsigned)(((size_t)N1c * H + 255) / 256), blk256, 0, stream>>>(
        ws + O_H1, ws + O_H1, ws + O_SS, (size_t)N1c * H, 1);

    // ---- P4: layer-1 edge messages + pooling ----
    gemm_wmma_f32<<<dim3(E1 / BM, H / BN), blk128, 0, stream>>>(
        ws + O_H1, W_pool1, H, nullptr, nullptr, 0,
        esrc1, b_pool1, ws + O_MEDGE1, H, 1);
    {
        int tot = N2c * (H / 4);
        segmax_kernel<<<(tot + 255) / 256, blk256, 0, stream>>>(
            ws + O_MEDGE1, ws + O_POOLED1, N2c, H, FAN1);
    }

    // ---- P5: h2 = h1[:N2]@W_self1 + pooled1@W_neigh1 + b1 ----
    gemm_wmma_f32<<<dim3(N2c / BM, H / BN), blk128, 0, stream>>>(
        ws + O_H1, W_self1, H, ws + O_POOLED1, W_neigh1, H,
        nullptr, b1, ws + O_H2, H, 0);

    // ---- P6: l2norm -> W_lat -> BNl -> d_out ----
    l2norm_kernel<<<N2c, dim3(H), 0, stream>>>(ws + O_H2, H);
    gemm_wmma_f32<<<dim3(N2c / BM, H / BN), blk128, 0, stream>>>(
        ws + O_H2, W_lat, H, nullptr, nullptr, 0, nullptr, b_lat, ws + O_HL, H, 0);
    colstat_partial<<<N2c / STAT_ROWS, blk256, 0, stream>>>(ws + O_HL, ws + O_PART, H);
    colstat_finalize<<<1, blk256, 0, stream>>>(ws + O_PART, N2c / STAT_ROWS,
                                               g_bnl, be_bnl, (float)N2c, H, ws + O_SS);
    bn_apply<<<(unsigned)(((size_t)N2c * H + 255) / 256), blk256, 0, stream>>>(
        ws + O_HL, (float*)d_out, ws + O_SS, (size_t)N2c * H, 0);
}